// GVAE_30648886624856
// MI455X (gfx1250) — compile-verified
//
#include <hip/hip_runtime.h>

// FP32 WMMA fragments (V_WMMA_F32_16X16X4_F32): A,B = 2 VGPRs, C/D = 8 VGPRs
typedef __attribute__((ext_vector_type(2))) float v2f;
typedef __attribute__((ext_vector_type(8))) float v8f;

#define NPG     32        // nodes per graph
#define EPG     256       // edges per graph
#define NGRAPH  1024
#define NNODES  32768
#define NEDGES  262144
#define NHEADS  3
#define EDIM    11
#define PPG     496       // pairs per graph (32 choose 2)
#define NPAIRS  507904
#define DECC    128
#define BN_EPS  1e-5f
#define PS      321       // LDS stride for proj (avoid 64-bank conflicts)

__device__ __forceinline__ v8f wmma4(v2f a, v2f b, v8f c) {
  // D = A(16x4,f32) * B(4x16,f32) + C(16x16,f32)
  return __builtin_amdgcn_wmma_f32_16x16x4_f32(false, a, false, b, (short)0, c,
                                               false, false);
}

// Order-preserving float <-> uint mapping so segment max can use ds_max_u32.
// Key 0 is reserved as "no edge" (smaller than any encoded float).
__device__ __forceinline__ unsigned fkey(float x) {
  unsigned u = __float_as_uint(x);
  return (u & 0x80000000u) ? ~u : (u | 0x80000000u);
}
__device__ __forceinline__ float funkey(unsigned k) {
  if (k == 0u) return 0.f;               // empty segment -> reference's isfinite fixup
  unsigned u = (k & 0x80000000u) ? (k & 0x7fffffffu) : ~k;
  return __uint_as_float(u);
}

// ---------------------------------------------------------------------------
// Pack one conv layer's weights into a contiguous K-padded matrix:
//   Wcat[k][col], k in [0,32) zero-padded past fin, col = [q|k|v|skip] (10c)
//   bcat[col] = packed bias row.
// Removes all conditionals from the WMMA inner loop of tconv_fused.
// ---------------------------------------------------------------------------
__global__ void pack_w(const float* __restrict__ Wq, const float* __restrict__ bq,
                       const float* __restrict__ Wk, const float* __restrict__ bk,
                       const float* __restrict__ Wv, const float* __restrict__ bv,
                       const float* __restrict__ Ws, const float* __restrict__ bs,
                       int fin, int c, float* __restrict__ Wcat, float* __restrict__ bcat)
{
  const int Hc = 3 * c, ncols = 10 * c;
  int idx = blockIdx.x * blockDim.x + threadIdx.x;
  int total = 32 * ncols;
  if (idx < total) {
    int k = idx / ncols, col = idx - k * ncols;
    const float* W; int ld, off;
    if (col < Hc)            { W = Wq; ld = Hc; off = col;          }
    else if (col < 2 * Hc)   { W = Wk; ld = Hc; off = col - Hc;     }
    else if (col < 3 * Hc)   { W = Wv; ld = Hc; off = col - 2 * Hc; }
    else                     { W = Ws; ld = c;  off = col - 3 * Hc; }
    Wcat[idx] = (k < fin) ? W[k * ld + off] : 0.f;
  } else if (idx < total + ncols) {
    int col = idx - total;
    const float* B; int off;
    if (col < Hc)            { B = bq; off = col;          }
    else if (col < 2 * Hc)   { B = bk; off = col - Hc;     }
    else if (col < 3 * Hc)   { B = bv; off = col - 2 * Hc; }
    else                     { B = bs; off = col - 3 * Hc; }
    bcat[col] = B[off];
  }
}

// ---------------------------------------------------------------------------
// Fused TransformerConv layer: one workgroup per graph (32 nodes, 256 edges).
//   proj = X @ Wcat + bcat via f32 WMMA (K=32, branch-free fragment loads)
//   edge projections edge_attr@We computed once per edge, kept in VGPRs
//   segment softmax via LDS atomics (ds_max_u32 on monotonic key, ds_add_f32)
//   beta gate in LDS; optional input BatchNorm + output ReLU + BN stats.
// ---------------------------------------------------------------------------
template <int C, int FIN>
__global__ __launch_bounds__(256)
void tconv_fused(const float* __restrict__ Xin,
                 const float* __restrict__ bn_s1, const float* __restrict__ bn_s2,
                 const float* __restrict__ bn_g,  const float* __restrict__ bn_b,
                 float invN,
                 const int* __restrict__ esrc, const int* __restrict__ edst,
                 const float* __restrict__ EA,
                 const float* __restrict__ Wcat, const float* __restrict__ bcat,
                 const float* __restrict__ We, const float* __restrict__ be,
                 const float* __restrict__ Wbeta,
                 float* __restrict__ Yout, int ldy, int do_relu,
                 float* __restrict__ stat_s1, float* __restrict__ stat_s2)
{
  constexpr int Hc    = NHEADS * C;
  constexpr int NCOLS = 10 * C;          // q|k|v|skip
  constexpr int NT    = NCOLS / 16;
  const int g    = blockIdx.x;
  const int tid  = threadIdx.x;
  const int lane = tid & 31;
  const int wave = __builtin_amdgcn_readfirstlane(tid >> 5);

  __shared__ float    xs[NPG * 33];       // node features, K padded to 32
  __shared__ float    proj[NPG * PS];     // [32][q|k|v|x_r]
  __shared__ unsigned nmaxU[NPG * NHEADS];// segment max keys
  __shared__ float    nmaxF[NPG * NHEADS];// decoded max (later: gate)
  __shared__ float    nden[NPG * NHEADS]; // segment denom
  __shared__ float    outacc[NPG * 96];   // attention output accumulator
  __shared__ float    csum[32], csq[32];  // per-channel BN partials

  // ---- phase 0: load X tile (optionally applying previous layer's BN) ----
  for (int idx = tid; idx < NPG * 32; idx += 256) {
    int row = idx >> 5, col = idx & 31;
    float v = 0.f;
    if (col < FIN) {
      v = Xin[(g * NPG + row) * FIN + col];
      if (bn_s1) {
        float mean = bn_s1[col] * invN;
        float var  = bn_s2[col] * invN - mean * mean;
        v = (v - mean) * rsqrtf(var + BN_EPS) * bn_g[col] + bn_b[col];
      }
    }
    xs[row * 33 + col] = v;
  }
  for (int idx = tid; idx < NPG * 96; idx += 256) outacc[idx] = 0.f;
  if (tid < NPG * NHEADS) { nmaxU[tid] = 0u; nden[tid] = 0.f; }
  if (tid < 32) { csum[tid] = 0.f; csq[tid] = 0.f; }
  __syncthreads();

  // ---- phase 1: proj = X @ Wcat + bcat  (WMMA f32, K=32, branch-free) ----
  {
    const int m  = lane & 15;             // A row / B col / C col
    const int kg = (lane >> 4) << 1;      // K group per lane half
    for (int t = wave; t < 2 * NT; t += 8) {
      const int mt  = t & 1;
      const int nc0 = (t >> 1) << 4;
      v8f acc = {};
      #pragma unroll
      for (int kk = 0; kk < 8; ++kk) {
        const int k0 = (kk << 2) + kg;
        v2f a, b;
        a.x = xs[(mt * 16 + m) * 33 + k0];
        a.y = xs[(mt * 16 + m) * 33 + k0 + 1];
        b.x = Wcat[(k0)     * NCOLS + nc0 + m];
        b.y = Wcat[(k0 + 1) * NCOLS + nc0 + m];
        acc = wmma4(a, b, acc);
      }
      const float bias = bcat[nc0 + m];
      const int rbase = mt * 16 + ((lane >> 4) << 3);
      #pragma unroll
      for (int r = 0; r < 8; ++r)
        proj[(rbase + r) * PS + nc0 + m] = acc[r] + bias;
    }
  }
  __syncthreads();

  // ---- phase 2: per-edge logits; edge proj kept in registers ----
  const int eid = g * EPG + tid;
  const int s = esrc[eid] & (NPG - 1);
  const int d = edst[eid] & (NPG - 1);
  float ea[EDIM];
  #pragma unroll
  for (int t = 0; t < EDIM; ++t) ea[t] = EA[eid * EDIM + t];

  float ef[NHEADS * C];                   // edge projection, reused in value pass
  float lg[NHEADS];
  {
    const float scl = rsqrtf((float)C);
    #pragma unroll
    for (int h = 0; h < NHEADS; ++h) {
      float acc = 0.f;
      #pragma unroll
      for (int j = 0; j < C; ++j) {
        int hc = h * C + j;
        float e = be[hc];
        #pragma unroll
        for (int t = 0; t < EDIM; ++t) e += ea[t] * We[t * Hc + hc];
        ef[hc] = e;
        acc += proj[d * PS + hc] * (proj[s * PS + Hc + hc] + e);
      }
      lg[h] = acc * scl;
      atomicMax(&nmaxU[h * NPG + d], fkey(lg[h]));   // segment max via ds_max_u32
    }
  }
  __syncthreads();

  // ---- phase 3: decode segment max ----
  if (tid < NPG * NHEADS) nmaxF[tid] = funkey(nmaxU[tid]);
  __syncthreads();

  // ---- phase 4: exp weights + segment denominators (ds_add_f32) ----
  float wv[NHEADS];
  #pragma unroll
  for (int h = 0; h < NHEADS; ++h) {
    wv[h] = expf(lg[h] - nmaxF[h * NPG + d]);
    atomicAdd(&nden[h * NPG + d], wv[h]);
  }
  __syncthreads();

  // ---- phase 5: weighted value scatter (edge proj from registers) ----
  #pragma unroll
  for (int h = 0; h < NHEADS; ++h) {
    float alpha = wv[h] / (nden[h * NPG + d] + 1e-16f);
    #pragma unroll
    for (int j = 0; j < C; ++j) {
      int hc = h * C + j;
      float vj = proj[s * PS + 2 * Hc + hc] + ef[hc];
      atomicAdd(&outacc[d * Hc + hc], alpha * vj);
    }
  }
  __syncthreads();

  // ---- phase 6a: head mean + beta gate (one thread per node) ----
  if (tid < NPG) {
    int nd = tid;
    float accg = 0.f;
    #pragma unroll
    for (int j = 0; j < C; ++j) {
      float o = (outacc[nd * Hc + j] + outacc[nd * Hc + C + j] +
                 outacc[nd * Hc + 2 * C + j]) * (1.f / 3.f);
      outacc[nd * Hc + j] = o;
      float xr = proj[nd * PS + 3 * Hc + j];
      accg += Wbeta[j] * o + Wbeta[C + j] * xr + Wbeta[2 * C + j] * (o - xr);
    }
    nmaxF[nd] = 1.f / (1.f + expf(-accg));   // reuse nmaxF as gate storage
  }
  __syncthreads();

  // ---- phase 6b: gated output (+ReLU), BN stat accumulation ----
  for (int idx = tid; idx < NPG * C; idx += 256) {
    int nd = idx / C;                      // C is constexpr -> shift
    int j  = idx - nd * C;
    float o  = outacc[nd * Hc + j];
    float xr = proj[nd * PS + 3 * Hc + j];
    float gt = nmaxF[nd];
    float res = gt * xr + (1.f - gt) * o;
    if (do_relu) res = fmaxf(res, 0.f);
    Yout[(g * NPG + nd) * ldy + j] = res;
    if (stat_s1) {
      atomicAdd(&csum[j], res);
      atomicAdd(&csq[j], res * res);
    }
  }
  __syncthreads();
  if (stat_s1 && tid < C) {
    atomicAdd(&stat_s1[tid], csum[tid]);
    atomicAdd(&stat_s2[tid], csq[tid]);
  }
}

// ---------------------------------------------------------------------------
__global__ void zero_f(float* p, int n) {
  int i = blockIdx.x * blockDim.x + threadIdx.x;
  if (i < n) p[i] = 0.f;
}

__global__ void reparam(const float* __restrict__ mu, const float* __restrict__ lv,
                        const float* __restrict__ eps, float* __restrict__ z, int n) {
  int i = blockIdx.x * blockDim.x + threadIdx.x;
  if (i < n) z[i] = eps[i] * expf(lv[i]) + mu[i];
}

__global__ void build_pairs(int* pi, int* pj) {
  int r = blockIdx.x * blockDim.x + threadIdx.x;
  if (r >= PPG) return;
  int i = 0, rem = r;
  while (rem >= NPG - 1 - i) { rem -= NPG - 1 - i; ++i; }
  pi[r] = i; pj[r] = i + 1 + rem;
}

// ---------------------------------------------------------------------------
// Decoder pass A: per-channel sums of relu(inp@W3+b3). One block per graph;
// pairs gather z from a 2KB LDS tile. WMMA f32, M=16 pairs, K=32, N=128.
// ---------------------------------------------------------------------------
__global__ __launch_bounds__(256)
void dec_stats(const float* __restrict__ z, const int* __restrict__ pi,
               const int* __restrict__ pj, const float* __restrict__ W3,
               const float* __restrict__ b3,
               float* __restrict__ dsum, float* __restrict__ dsq)
{
  const int g = blockIdx.x, tid = threadIdx.x;
  const int lane = tid & 31;
  const int wave = __builtin_amdgcn_readfirstlane(tid >> 5);
  __shared__ float zg[NPG * 17];
  __shared__ float csum[DECC], csq[DECC];
  for (int idx = tid; idx < NPG * 16; idx += 256) {
    int r = idx >> 4, c0 = idx & 15;
    zg[r * 17 + c0] = z[(g * NPG + r) * 16 + c0];
  }
  if (tid < DECC) { csum[tid] = 0.f; csq[tid] = 0.f; }
  __syncthreads();
  const int m  = lane & 15;
  const int kg = (lane >> 4) << 1;
  for (int rg = wave; rg < 31; rg += 8) {          // 31 row-groups of 16 pairs
    int p  = rg * 16 + m;
    int ia = pi[p], ja = pj[p];
    #pragma unroll
    for (int nt = 0; nt < 8; ++nt) {
      v8f acc = {};
      #pragma unroll
      for (int kk = 0; kk < 8; ++kk) {
        int k0 = (kk << 2) + kg;
        v2f a, b;
        a.x = (k0 < 16) ? zg[ia * 17 + k0] : zg[ja * 17 + k0 - 16];
        a.y = (k0 < 16) ? zg[ia * 17 + k0 + 1] : zg[ja * 17 + k0 - 15];
        b.x = W3[(k0)     * DECC + nt * 16 + m];
        b.y = W3[(k0 + 1) * DECC + nt * 16 + m];
        acc = wmma4(a, b, acc);
      }
      int col = nt * 16 + m;
      float bb = b3[col];
      float s1 = 0.f, s2 = 0.f;
      #pragma unroll
      for (int r = 0; r < 8; ++r) {
        float h = fmaxf(acc[r] + bb, 0.f);
        s1 += h; s2 += h * h;
      }
      atomicAdd(&csum[col], s1);
      atomicAdd(&csq[col],  s2);
    }
  }
  __syncthreads();
  if (tid < DECC) { atomicAdd(&dsum[tid], csum[tid]); atomicAdd(&dsq[tid], csq[tid]); }
}

// Fold decoder BN + W4 into an effective weight vector and scalar:
// logit = sum_c h_c * weff_c + coeff
__global__ void dec_weff(const float* __restrict__ dsum, const float* __restrict__ dsq,
                         const float* __restrict__ g3, const float* __restrict__ be3,
                         const float* __restrict__ W4, const float* __restrict__ b4,
                         float invP, float* __restrict__ weff, float* __restrict__ coeff)
{
  int t = threadIdx.x;   // 128
  float mean = dsum[t] * invP;
  float var  = dsq[t] * invP - mean * mean;
  float inv  = rsqrtf(var + BN_EPS);
  weff[t] = g3[t] * inv * W4[t];
  __shared__ float red[DECC];
  red[t] = (be3[t] - mean * g3[t] * inv) * W4[t];
  __syncthreads();
  for (int s = 64; s > 0; s >>= 1) { if (t < s) red[t] += red[t + s]; __syncthreads(); }
  if (t == 0) coeff[0] = red[0] + b4[0];
}

// Decoder pass B: recompute GEMM, emit logits.
__global__ __launch_bounds__(256)
void dec_logits(const float* __restrict__ z, const int* __restrict__ pi,
                const int* __restrict__ pj, const float* __restrict__ W3,
                const float* __restrict__ b3, const float* __restrict__ weff,
                const float* __restrict__ coeff, float* __restrict__ out)
{
  const int g = blockIdx.x, tid = threadIdx.x;
  const int lane = tid & 31;
  const int wave = __builtin_amdgcn_readfirstlane(tid >> 5);
  __shared__ float zg[NPG * 17];
  for (int idx = tid; idx < NPG * 16; idx += 256) {
    int r = idx >> 4, c0 = idx & 15;
    zg[r * 17 + c0] = z[(g * NPG + r) * 16 + c0];
  }
  __syncthreads();
  const float cf = coeff[0];
  const int m  = lane & 15;
  const int kg = (lane >> 4) << 1;
  for (int rg = wave; rg < 31; rg += 8) {
    int p  = rg * 16 + m;
    int ia = pi[p], ja = pj[p];
    float accw[8] = {0.f, 0.f, 0.f, 0.f, 0.f, 0.f, 0.f, 0.f};
    #pragma unroll
    for (int nt = 0; nt < 8; ++nt) {
      v8f acc = {};
      #pragma unroll
      for (int kk = 0; kk < 8; ++kk) {
        int k0 = (kk << 2) + kg;
        v2f a, b;
        a.x = (k0 < 16) ? zg[ia * 17 + k0] : zg[ja * 17 + k0 - 16];
        a.y = (k0 < 16) ? zg[ia * 17 + k0 + 1] : zg[ja * 17 + k0 - 15];
        b.x = W3[(k0)     * DECC + nt * 16 + m];
        b.y = W3[(k0 + 1) * DECC + nt * 16 + m];
        acc = wmma4(a, b, acc);
      }
      int col = nt * 16 + m;
      float bb = b3[col];
      float wf = weff[col];
      #pragma unroll
      for (int r = 0; r < 8; ++r)
        accw[r] += fmaxf(acc[r] + bb, 0.f) * wf;
    }
    #pragma unroll
    for (int r = 0; r < 8; ++r) {           // reduce 16 cols per lane-half
      float v = accw[r];
      v += __shfl_xor(v, 1, 32);
      v += __shfl_xor(v, 2, 32);
      v += __shfl_xor(v, 4, 32);
      v += __shfl_xor(v, 8, 32);
      accw[r] = v;
    }
    if (m == 0) {
      int rowoff = (lane >> 4) << 3;
      #pragma unroll
      for (int r = 0; r < 8; ++r)
        out[g * PPG + rg * 16 + rowoff + r] = accw[r] + cf;
    }
  }
}

// ---------------------------------------------------------------------------
// Input index map assumption: top-level setup_inputs() insertion order, with
// the nested 'params' pytree flattened jax-style (sorted dict keys):
//  0:x 1:edge_attr 2:eps
//  3:bn1_b 4:bn1_g 5:bn2_b 6:bn2_g 7:bn3_b 8:bn3_g
//  9..19:conv1  20..30:conv2  31..41:conv3   (Wbeta,We,Wk,Wq,Ws,Wv,be,bk,bq,bs,bv)
//  42..55:dec (W1,W2,W3,W4,b1,b2,b3,b4,be1,be2,be3,g1,g2,g3)
//  56..66:logvar conv   67..77:mu conv
//  78:edge_index  79:nodes_per_graph
// ---------------------------------------------------------------------------
struct CP { const float *Wq,*bq,*Wk,*bk,*Wv,*bv,*We,*be,*Ws,*bs,*Wbeta; };
static CP conv_params(void* const* d_in, int base) {
  CP p;
  p.Wbeta = (const float*)d_in[base + 0];
  p.We    = (const float*)d_in[base + 1];
  p.Wk    = (const float*)d_in[base + 2];
  p.Wq    = (const float*)d_in[base + 3];
  p.Ws    = (const float*)d_in[base + 4];
  p.Wv    = (const float*)d_in[base + 5];
  p.be    = (const float*)d_in[base + 6];
  p.bk    = (const float*)d_in[base + 7];
  p.bq    = (const float*)d_in[base + 8];
  p.bs    = (const float*)d_in[base + 9];
  p.bv    = (const float*)d_in[base + 10];
  return p;
}

extern "C" void kernel_launch(void* const* d_in, const int* in_sizes, int n_in,
                              void* d_out, int out_size, void* d_ws, size_t ws_size,
                              hipStream_t stream) {
  (void)in_sizes; (void)out_size; (void)ws_size;
  const float* x   = (const float*)d_in[0];
  const float* ea  = (const float*)d_in[1];
  const float* eps = (const float*)d_in[2];
  const int* eidx  = (const int*)d_in[n_in - 2];   // edge_index [2,E]
  const int* esrc  = eidx;
  const int* edst  = eidx + NEDGES;

  CP c1 = conv_params(d_in, 9),  c2 = conv_params(d_in, 20);
  CP c3 = conv_params(d_in, 31), clv = conv_params(d_in, 56), cmu = conv_params(d_in, 67);
  const float* bn1_b = (const float*)d_in[3]; const float* bn1_g = (const float*)d_in[4];
  const float* bn2_b = (const float*)d_in[5]; const float* bn2_g = (const float*)d_in[6];
  const float* bn3_b = (const float*)d_in[7]; const float* bn3_g = (const float*)d_in[8];
  const float* W3  = (const float*)d_in[44]; const float* W4  = (const float*)d_in[45];
  const float* b3  = (const float*)d_in[48]; const float* b4  = (const float*)d_in[49];
  const float* be3 = (const float*)d_in[52]; const float* g3  = (const float*)d_in[55];

  float* out        = (float*)d_out;
  float* out_logits = out;
  float* out_mu     = out + NPAIRS;
  float* out_lv     = out + NPAIRS + NNODES * 16;

  float* ws    = (float*)d_ws;
  float* h_a   = ws;                           // [32768*32]
  float* h_b   = h_a + NNODES * 32;            // [32768*32]
  float* zbuf  = h_b + NNODES * 32;            // [32768*16]
  float* stats = zbuf + NNODES * 16;           // 6*32 + 2*128 + 128 + 1
  float* s1l1 = stats +   0; float* s2l1 = stats +  32;
  float* s1l2 = stats +  64; float* s2l2 = stats +  96;
  float* s1l3 = stats + 128; float* s2l3 = stats + 160;
  float* dsum = stats + 192; float* dsq  = dsum + 128;
  float* weff = dsq + 128;   float* coeff = weff + 128;
  int*   pi   = (int*)(coeff + 1);
  int*   pj   = pi + PPG;
  // packed weight matrices (K padded to 32) + packed bias rows
  float* wc1 = (float*)(pj + PPG);             // 32*320 + 320
  float* bc1 = wc1 + 32 * 320;
  float* wc2 = bc1 + 320;  float* bc2 = wc2 + 32 * 320;
  float* wc3 = bc2 + 320;  float* bc3 = wc3 + 32 * 320;
  float* wcm = bc3 + 320;  float* bcm = wcm + 32 * 160;   // mu
  float* wcl = bcm + 160;  float* bcl = wcl + 32 * 160;   // logvar

  const float invN = 1.f / (float)NNODES;
  const int PACK_ENC = (32 * 320 + 320 + 255) / 256;
  const int PACK_LAT = (32 * 160 + 160 + 255) / 256;

  zero_f<<<2, 256, 0, stream>>>(stats, 192 + 256);
  build_pairs<<<2, 256, 0, stream>>>(pi, pj);
  pack_w<<<PACK_ENC, 256, 0, stream>>>(c1.Wq, c1.bq, c1.Wk, c1.bk, c1.Wv, c1.bv,
                                       c1.Ws, c1.bs, 30, 32, wc1, bc1);
  pack_w<<<PACK_ENC, 256, 0, stream>>>(c2.Wq, c2.bq, c2.Wk, c2.bk, c2.Wv, c2.bv,
                                       c2.Ws, c2.bs, 32, 32, wc2, bc2);
  pack_w<<<PACK_ENC, 256, 0, stream>>>(c3.Wq, c3.bq, c3.Wk, c3.bk, c3.Wv, c3.bv,
                                       c3.Ws, c3.bs, 32, 32, wc3, bc3);
  pack_w<<<PACK_LAT, 256, 0, stream>>>(cmu.Wq, cmu.bq, cmu.Wk, cmu.bk, cmu.Wv, cmu.bv,
                                       cmu.Ws, cmu.bs, 32, 16, wcm, bcm);
  pack_w<<<PACK_LAT, 256, 0, stream>>>(clv.Wq, clv.bq, clv.Wk, clv.bk, clv.Wv, clv.bv,
                                       clv.Ws, clv.bs, 32, 16, wcl, bcl);

  // encoder
  tconv_fused<32, 30><<<NGRAPH, 256, 0, stream>>>(x, nullptr, nullptr, nullptr, nullptr,
      invN, esrc, edst, ea, wc1, bc1, c1.We, c1.be, c1.Wbeta, h_a, 32, 1, s1l1, s2l1);
  tconv_fused<32, 32><<<NGRAPH, 256, 0, stream>>>(h_a, s1l1, s2l1, bn1_g, bn1_b,
      invN, esrc, edst, ea, wc2, bc2, c2.We, c2.be, c2.Wbeta, h_b, 32, 1, s1l2, s2l2);
  tconv_fused<32, 32><<<NGRAPH, 256, 0, stream>>>(h_b, s1l2, s2l2, bn2_g, bn2_b,
      invN, esrc, edst, ea, wc3, bc3, c3.We, c3.be, c3.Wbeta, h_a, 32, 1, s1l3, s2l3);
  tconv_fused<16, 32><<<NGRAPH, 256, 0, stream>>>(h_a, s1l3, s2l3, bn3_g, bn3_b,
      invN, esrc, edst, ea, wcm, bcm, cmu.We, cmu.be, cmu.Wbeta, out_mu, 16, 0,
      nullptr, nullptr);
  tconv_fused<16, 32><<<NGRAPH, 256, 0, stream>>>(h_a, s1l3, s2l3, bn3_g, bn3_b,
      invN, esrc, edst, ea, wcl, bcl, clv.We, clv.be, clv.Wbeta, out_lv, 16, 0,
      nullptr, nullptr);

  // reparameterize + decoder
  reparam<<<(NNODES * 16 + 255) / 256, 256, 0, stream>>>(out_mu, out_lv, eps, zbuf, NNODES * 16);
  dec_stats<<<NGRAPH, 256, 0, stream>>>(zbuf, pi, pj, W3, b3, dsum, dsq);
  dec_weff<<<1, 128, 0, stream>>>(dsum, dsq, g3, be3, W4, b4, 1.f / (float)NPAIRS, weff, coeff);
  dec_logits<<<NGRAPH, 256, 0, stream>>>(zbuf, pi, pj, W3, b3, weff, coeff, out_logits);
}